// Explainer_70755291234400
// MI455X (gfx1250) — compile-verified
//
#include <hip/hip_runtime.h>
#include <hip/hip_bf16.h>

typedef __attribute__((ext_vector_type(2))) float v2f;
typedef __attribute__((ext_vector_type(8))) float v8f;

#define BATCH 2048
#define KDIM  32
#define DDIM  256
#define NCOLS (DDIM * DDIM)       // 65536 columns
#define MTILES (BATCH / 16)       // 128 row tiles
#define NTILES (NCOLS / 16)       // 4096 column tiles
#define WAVES_PER_BLOCK 8

// C(2048 x 65536) = W(2048 x 32) * A(32 x 65536), column n zeroed iff n % 257 == 0.
// One wave per 16-wide column tile; B tile cached in VGPRs, loop over 128 M tiles.
__global__ void __launch_bounds__(WAVES_PER_BLOCK * 32)
explainer_wmma_f32_kernel(const float* __restrict__ W,
                          const float* __restrict__ A,
                          float* __restrict__ out) {
    const int lane = threadIdx.x & 31;
    const int h    = lane >> 4;        // lane half: 0 or 1
    const int l16  = lane & 15;
    const int waveInBlock = threadIdx.x >> 5;
    const int bn = blockIdx.x * WAVES_PER_BLOCK + waveInBlock;  // column tile id
    if (bn >= NTILES) return;

    const int col = bn * 16 + l16;                 // global column n
    // diagonal mask: n = d*256+e is on-diagonal iff n % 257 == 0
    const float dmask = (col % 257 == 0) ? 0.0f : 1.0f;

    // ---- Load B tile (32 x 16) of archs, masked, into 8 x v2f registers ----
    // B layout (4x16 per step): VGPR comp v, lane half h -> K = 4s + 2h + v, N = l16
    v2f b[8];
#pragma unroll
    for (int s = 0; s < 8; ++s) {
        const int k0 = 4 * s + 2 * h;
        b[s].x = A[(size_t)k0 * NCOLS + col] * dmask;
        b[s].y = A[(size_t)(k0 + 1) * NCOLS + col] * dmask;
    }

    // ---- Loop over 128 batch (M) tiles ----
    for (int bm = 0; bm < MTILES; ++bm) {
        // A layout (16x4 per step): lane holds row M = l16; comp v, half h -> K = 4s + 2h + v
        const float* wrow = W + (size_t)(bm * 16 + l16) * KDIM;
        v8f c = {};
#pragma unroll
        for (int s = 0; s < 8; ++s) {
            const int k0 = 4 * s + 2 * h;          // even -> 8B aligned b64 load
            v2f a = *(const v2f*)(wrow + k0);
            c = __builtin_amdgcn_wmma_f32_16x16x4_f32(
                    /*neg_a=*/false, a, /*neg_b=*/false, b[s],
                    /*c_mod=*/(short)0, c, /*reuse_a=*/false, /*reuse_b=*/false);
        }
        // ---- Store C tile: VGPR v, half h -> row M = v + 8h, col N = l16 ----
        // Non-temporal: 512 MB streamed output must not evict L2-resident A/W.
        float* orow = out + (size_t)(bm * 16) * NCOLS + col;
#pragma unroll
        for (int v = 0; v < 8; ++v) {
            const int m = v + 8 * h;
            __builtin_nontemporal_store(c[v], orow + (size_t)m * NCOLS);
        }
    }
}

extern "C" void kernel_launch(void* const* d_in, const int* in_sizes, int n_in,
                              void* d_out, int out_size, void* d_ws, size_t ws_size,
                              hipStream_t stream) {
    const float* W = (const float*)d_in[0];   // (2048, 32) f32
    const float* A = (const float*)d_in[1];   // (32, 256, 256) f32
    float* out = (float*)d_out;               // (2048, 256, 256) f32

    const int blocks = NTILES / WAVES_PER_BLOCK;   // 512
    explainer_wmma_f32_kernel<<<blocks, WAVES_PER_BLOCK * 32, 0, stream>>>(W, A, out);
}